// VariationalQuantumEigensolver_1786706395147
// MI455X (gfx1250) — compile-verified
//
#include <hip/hip_runtime.h>
#include <hip/hip_bf16.h>
#include <math.h>

// ---------------------------------------------------------------------------
// Problem constants (from reference)
// ---------------------------------------------------------------------------
#define NQ    16
#define NL    3
#define BATCH 32
#define NSTATE 65536            // 2^NQ amplitudes (real-valued, see analysis)
#define OUT_ROW 131328          // 2*65536 + 256 per batch row

typedef float v2f __attribute__((ext_vector_type(2)));
typedef float v8f __attribute__((ext_vector_type(8)));

// CDNA5 fp32 WMMA: D(16x16) += A(16x4) * B(4x16), full fp32.
__device__ __forceinline__ v8f wmma_f32_k4(v2f a, v2f b, v8f c) {
    // 8 args: (neg_a, A, neg_b, B, c_mod, C, reuse_a, reuse_b)
    return __builtin_amdgcn_wmma_f32_16x16x4_f32(
        false, a, false, b, (short)0, c, false, false);
}

// ---------------------------------------------------------------------------
// Kernel 1: feature-map MLP  pf = tanh(relu(feat @ w1^T + b1) @ w2^T + b2)
//   feat (32,100), w1 (64,100), w2 (16,64)  ->  pf (32,16) in workspace
//   One workgroup, 8 waves; each wave owns one 16x16 WMMA tile of layer 1.
// ---------------------------------------------------------------------------
__global__ __launch_bounds__(256) void feature_mlp_kernel(
    const float* __restrict__ feat,
    const float* __restrict__ w1, const float* __restrict__ b1,
    const float* __restrict__ w2, const float* __restrict__ b2,
    float* __restrict__ pf)
{
    __shared__ float fs[BATCH * 100];   // features
    __shared__ float h [BATCH * 64];    // hidden

    const int tid  = threadIdx.x;
    const int wave = tid >> 5;
    const int lane = tid & 31;
    const int half = lane >> 4;         // 0: K even pair, 1: K odd pair
    const int l    = lane & 15;         // row-in-tile (A) / col-in-tile (B,D)

    for (int i = tid; i < BATCH * 100; i += 256) fs[i] = feat[i];
    __syncthreads();

    // ---- layer 1: (32x100) x (100x64), tiles: Mt=2 x Nt=4 == 8 waves ----
    {
        const int m0 = (wave & 1) * 16;
        const int n0 = (wave >> 1) * 16;
        v8f c = {};
        for (int k0 = 0; k0 < 100; k0 += 4) {
            const int k = k0 + 2 * half;
            v2f a, bf;
            a.x  = fs[(m0 + l) * 100 + k];
            a.y  = fs[(m0 + l) * 100 + k + 1];
            bf.x = w1[(n0 + l) * 100 + k];       // B[k][n] = w1[n][k]
            bf.y = w1[(n0 + l) * 100 + k + 1];
            c = wmma_f32_k4(a, bf, c);
        }
        #pragma unroll
        for (int r = 0; r < 8; ++r) {
            const int m = m0 + 8 * half + r;
            const int n = n0 + l;
            const float v = c[r] + b1[n];
            h[m * 64 + n] = v > 0.f ? v : 0.f;
        }
    }
    __syncthreads();

    // ---- layer 2: (32x64) x (64x16), tiles: Mt=2 x Nt=1 -> waves 0,1 ----
    if (wave < 2) {
        const int m0 = wave * 16;
        v8f c = {};
        for (int k0 = 0; k0 < 64; k0 += 4) {
            const int k = k0 + 2 * half;
            v2f a, bf;
            a.x  = h[(m0 + l) * 64 + k];
            a.y  = h[(m0 + l) * 64 + k + 1];
            bf.x = w2[l * 64 + k];
            bf.y = w2[l * 64 + k + 1];
            c = wmma_f32_k4(a, bf, c);
        }
        #pragma unroll
        for (int r = 0; r < 8; ++r) {
            const int m = m0 + 8 * half + r;
            pf[m * NQ + l] = tanhf(c[r] + b2[l]);
        }
    }
}

// ---------------------------------------------------------------------------
// Kernel 2: Hamiltonian MLP + symmetrize + freq*I, writes H block of output.
//   x (32,60) -> relu(x w1^T) (32,256) -> relu(. w2^T) (32,128) -> . w3^T (32,256)
// ---------------------------------------------------------------------------
__global__ __launch_bounds__(256) void hamiltonian_kernel(
    const float* __restrict__ coords,                       // (32,60)
    const float* __restrict__ w1, const float* __restrict__ b1,  // (256,60)
    const float* __restrict__ w2, const float* __restrict__ b2,  // (128,256)
    const float* __restrict__ w3, const float* __restrict__ b3,  // (256,128)
    float* __restrict__ out)
{
    __shared__ float xs[BATCH * 60];
    __shared__ float h1[BATCH * 256];   // layer-1 act, later reused for h3
    __shared__ float h2[BATCH * 128];
    __shared__ float freq[BATCH];

    const int tid  = threadIdx.x;
    const int wave = tid >> 5;
    const int lane = tid & 31;
    const int half = lane >> 4;
    const int l    = lane & 15;

    for (int i = tid; i < BATCH * 60; i += 256) xs[i] = coords[i];
    __syncthreads();

    // freq[b] = sqrt(1/(var+1e-6))*200, var with ddof=1 over 60 coords
    if (tid < BATCH) {
        float s = 0.f;
        for (int i = 0; i < 60; ++i) s += xs[tid * 60 + i];
        const float mean = s * (1.f / 60.f);
        float v = 0.f;
        for (int i = 0; i < 60; ++i) {
            const float d = xs[tid * 60 + i] - mean;
            v += d * d;
        }
        v *= (1.f / 59.f);
        freq[tid] = sqrtf(1.f / (v + 1e-6f)) * 200.f;
    }

    // ---- layer 1: (32x60)x(60x256): 32 tiles over 8 waves, K=60 ----
    for (int t = wave; t < 32; t += 8) {
        const int m0 = (t & 1) * 16;
        const int n0 = (t >> 1) * 16;
        v8f c = {};
        for (int k0 = 0; k0 < 60; k0 += 4) {
            const int k = k0 + 2 * half;
            v2f a, bf;
            a.x  = xs[(m0 + l) * 60 + k];
            a.y  = xs[(m0 + l) * 60 + k + 1];
            bf.x = w1[(n0 + l) * 60 + k];
            bf.y = w1[(n0 + l) * 60 + k + 1];
            c = wmma_f32_k4(a, bf, c);
        }
        #pragma unroll
        for (int r = 0; r < 8; ++r) {
            const int m = m0 + 8 * half + r;
            const int n = n0 + l;
            const float v = c[r] + b1[n];
            h1[m * 256 + n] = v > 0.f ? v : 0.f;
        }
    }
    __syncthreads();

    // ---- layer 2: (32x256)x(256x128): 16 tiles, K=256 ----
    for (int t = wave; t < 16; t += 8) {
        const int m0 = (t & 1) * 16;
        const int n0 = (t >> 1) * 16;
        v8f c = {};
        for (int k0 = 0; k0 < 256; k0 += 4) {
            const int k = k0 + 2 * half;
            v2f a, bf;
            a.x  = h1[(m0 + l) * 256 + k];
            a.y  = h1[(m0 + l) * 256 + k + 1];
            bf.x = w2[(n0 + l) * 256 + k];
            bf.y = w2[(n0 + l) * 256 + k + 1];
            c = wmma_f32_k4(a, bf, c);
        }
        #pragma unroll
        for (int r = 0; r < 8; ++r) {
            const int m = m0 + 8 * half + r;
            const int n = n0 + l;
            const float v = c[r] + b2[n];
            h2[m * 128 + n] = v > 0.f ? v : 0.f;
        }
    }
    __syncthreads();

    // ---- layer 3: (32x128)x(128x256) -> h3 stored into h1's LDS ----
    for (int t = wave; t < 32; t += 8) {
        const int m0 = (t & 1) * 16;
        const int n0 = (t >> 1) * 16;
        v8f c = {};
        for (int k0 = 0; k0 < 128; k0 += 4) {
            const int k = k0 + 2 * half;
            v2f a, bf;
            a.x  = h2[(m0 + l) * 128 + k];
            a.y  = h2[(m0 + l) * 128 + k + 1];
            bf.x = w3[(n0 + l) * 128 + k];
            bf.y = w3[(n0 + l) * 128 + k + 1];
            c = wmma_f32_k4(a, bf, c);
        }
        #pragma unroll
        for (int r = 0; r < 8; ++r) {
            const int m = m0 + 8 * half + r;
            const int n = n0 + l;
            h1[m * 256 + n] = c[r] + b3[n];
        }
    }
    __syncthreads();

    // ---- epilogue: H = (h3 + h3^T)/2 + freq*I -> out[b][131072 + i*16 + j]
    for (int i = tid; i < BATCH * 256; i += 256) {
        const int b   = i >> 8;
        const int idx = i & 255;
        const int r   = idx >> 4;
        const int cc  = idx & 15;
        float v = 0.5f * (h1[b * 256 + idx] + h1[b * 256 + (cc * 16 + r)]);
        if (r == cc) v += freq[b];
        out[(size_t)b * OUT_ROW + 2 * NSTATE + idx] = v;
    }
}

// ---------------------------------------------------------------------------
// Kernel 3: quantum state evolution, one workgroup per batch element.
//   The 2^16 REAL amplitudes (R and CNOT coefficients are real; imag == 0)
//   live entirely in CDNA5's 320KB WGP LDS: 93 gate sweeps with no HBM
//   traffic. All sweeps use float2/float4 LDS accesses (ds b64/b128) to
//   halve the DS instruction issue versus scalar pair updates.
// ---------------------------------------------------------------------------
__global__ __launch_bounds__(1024) void vqe_state_kernel(
    const float* __restrict__ rot,   // (3,16,3)
    const float* __restrict__ ent,   // (3,15)
    const float* __restrict__ pf,    // (32,16) from workspace
    float* __restrict__ out)         // (32, 131328)
{
    extern __shared__ float amps[];  // 65536 floats = 256 KB
    const int b   = blockIdx.x;
    const int tid = threadIdx.x;     // blockDim.x == 1024 (32 wave32s)

    // vectorized zero-fill (ds_store_b128)
    {
        const float4 z = make_float4(0.f, 0.f, 0.f, 0.f);
        float4* a4 = (float4*)amps;
        for (int i = tid; i < (NSTATE >> 2); i += 1024) a4[i] = z;
    }
    __syncthreads();
    if (tid == 0) amps[0] = 1.f;
    __syncthreads();

    for (int layer = 0; layer < NL; ++layer) {
        // ---- 16 single-qubit rotations (real 2x2 matrices) ----
        for (int q = 0; q < NQ; ++q) {
            const float f  = pf[b * NQ + q];
            const float rx = rot[(layer * NQ + q) * 3 + 0] * f * 0.5f;
            const float ry = rot[(layer * NQ + q) * 3 + 1] * f * 0.5f;
            const float rz = rot[(layer * NQ + q) * 3 + 2] * f * 0.5f;
            const float cx = cosf(rx), sx = sinf(rx);
            const float cy = cosf(ry), sy = sinf(ry);
            const float cz = cosf(rz), sz = sinf(rz);
            const float r00 =  cx * cy * cz, r01 = -sx * sy * sz;
            const float r10 =  sx * sy * cz, r11 =  cx * cy * sz;

            if (q == 0) {
                // pair elements are adjacent: one b64 load + one b64 store
                for (int p = tid; p < (NSTATE >> 1); p += 1024) {
                    float2* ap = (float2*)(amps + (p << 1));
                    const float2 a = *ap;
                    float2 r;
                    r.x = r00 * a.x + r01 * a.y;
                    r.y = r10 * a.x + r11 * a.y;
                    *ap = r;
                }
            } else {
                // two adjacent pairs per iteration (p even): float2 halves
                const int st  = 1 << q;
                const int msk = st - 1;
                for (int p2 = tid; p2 < (NSTATE >> 2); p2 += 1024) {
                    const int p  = p2 << 1;
                    const int i0 = ((p & ~msk) << 1) | (p & msk);   // even
                    float2* a0p = (float2*)(amps + i0);
                    float2* a1p = (float2*)(amps + (i0 | st));
                    const float2 a0 = *a0p;
                    const float2 a1 = *a1p;
                    float2 n0v, n1v;
                    n0v.x = r00 * a0.x + r01 * a1.x;
                    n0v.y = r00 * a0.y + r01 * a1.y;
                    n1v.x = r10 * a0.x + r11 * a1.x;
                    n1v.y = r10 * a0.y + r11 * a1.y;
                    *a0p = n0v;
                    *a1p = n1v;
                }
            }
            __syncthreads();
        }
        // ---- 15 soft CNOTs: mix bit (c+1) where bit c == 1 ----
        for (int c = 0; c < NQ - 1; ++c) {
            const float pe = 1.f / (1.f + expf(-ent[layer * (NQ - 1) + c]));
            const float qe = 1.f - pe;

            if (c == 0) {
                // quad 4h..4h+3 contiguous: one b128 load + one b128 store
                float4* a4 = (float4*)amps;
                for (int h = tid; h < (NSTATE >> 2); h += 1024) {
                    float4 v = a4[h];
                    const float t0 = v.y;      // index 4h+1 (bit0=1, bit1=0)
                    const float t1 = v.w;      // index 4h+3 (bit0=1, bit1=1)
                    v.y = qe * t0 + pe * t1;
                    v.w = qe * t1 + pe * t0;
                    a4[h] = v;
                }
            } else {
                // two adjacent pairs per iteration: float2 halves
                const int msk = (1 << c) - 1;
                for (int p2 = tid; p2 < (NSTATE >> 3); p2 += 1024) {
                    const int p  = p2 << 1;
                    const int h  = p >> c;
                    const int lo = p & msk;                          // even
                    const int i0 = (h << (c + 2)) | (1 << c) | lo;   // even
                    float2* t0p = (float2*)(amps + i0);
                    float2* t1p = (float2*)(amps + (i0 | (1 << (c + 1))));
                    const float2 t0 = *t0p;
                    const float2 t1 = *t1p;
                    float2 n0v, n1v;
                    n0v.x = qe * t0.x + pe * t1.x;
                    n0v.y = qe * t0.y + pe * t1.y;
                    n1v.x = qe * t1.x + pe * t0.x;
                    n1v.y = qe * t1.y + pe * t0.y;
                    *t0p = n0v;
                    *t1p = n1v;
                }
            }
            __syncthreads();
        }
    }

    // real part = state, imag part = exact zeros (b128 stores throughout)
    {
        float* ob = out + (size_t)b * OUT_ROW;
        const float4* a4 = (const float4*)amps;
        float4* or4 = (float4*)ob;
        float4* oi4 = (float4*)(ob + NSTATE);
        const float4 z = make_float4(0.f, 0.f, 0.f, 0.f);
        for (int i = tid; i < (NSTATE >> 2); i += 1024) {
            or4[i] = a4[i];
            oi4[i] = z;
        }
    }
}

// ---------------------------------------------------------------------------
extern "C" void kernel_launch(void* const* d_in, const int* in_sizes, int n_in,
                              void* d_out, int out_size, void* d_ws, size_t ws_size,
                              hipStream_t stream) {
    const float* coords = (const float*)d_in[0];   // (32,20,3) -> (32,60)
    const float* feats  = (const float*)d_in[1];   // (32,100)
    const float* rot    = (const float*)d_in[2];   // (3,16,3)
    const float* ent    = (const float*)d_in[3];   // (3,15)
    const float* fp_w1  = (const float*)d_in[4];
    const float* fp_b1  = (const float*)d_in[5];
    const float* fp_w2  = (const float*)d_in[6];
    const float* fp_b2  = (const float*)d_in[7];
    const float* hc_w1  = (const float*)d_in[8];
    const float* hc_b1  = (const float*)d_in[9];
    const float* hc_w2  = (const float*)d_in[10];
    const float* hc_b2  = (const float*)d_in[11];
    const float* hc_w3  = (const float*)d_in[12];
    const float* hc_b3  = (const float*)d_in[13];

    float* out = (float*)d_out;
    float* pf  = (float*)d_ws;                     // (32,16) scratch

    feature_mlp_kernel<<<1, 256, 0, stream>>>(feats, fp_w1, fp_b1, fp_w2, fp_b2, pf);
    hamiltonian_kernel<<<1, 256, 0, stream>>>(coords, hc_w1, hc_b1, hc_w2, hc_b2,
                                              hc_w3, hc_b3, out);
    vqe_state_kernel<<<BATCH, 1024, NSTATE * sizeof(float), stream>>>(rot, ent, pf, out);
}